// FeatureContrastMasking_30142080483670
// MI455X (gfx1250) — compile-verified
//
#include <hip/hip_runtime.h>
#include <math.h>

// FeatureContrastMasking on [16,64,256,256] f32.
// Three separable 13-tap depthwise filters as banded-Toeplitz matmuls on
// V_WMMA_F32_16X16X4_F32 (full f32 precision), fused pointwise math.
// One wave per 16x32 output tile; 4 waves / block.
// Tile coordinates are forced into SGPRs (readfirstlane) so all edge-handling
// branches are scalar and EXEC stays all-ones around every WMMA (ISA rule).

typedef float v2f __attribute__((ext_vector_type(2)));
typedef float v4f __attribute__((ext_vector_type(4)));
typedef float v8f __attribute__((ext_vector_type(8)));

#define IMG_H 256
#define IMG_W 256
#define TILE_W 16
#define TILE_H 32
#define PADT 6                 // top/bottom halo rows
#define PADL 8                 // left halo cols (chosen for 16B alignment)
#define PATCH_H 44             // TILE_H + 12
#define PSTR 36                // patch row stride (floats, 16B multiple, bank-spread)
#define CSTR 44                // column-major intermediate: floats per column
#define NWAVES 4               // waves per block (128 threads)

__device__ __forceinline__ int refl256(int i) {
    // jnp.pad 'reflect' (edge not repeated): -k -> k, 255+k -> 255-k
    i = (i < 0) ? -i : i;
    return (i > 255) ? (510 - i) : i;
}

__device__ __forceinline__ v8f wmma4(v2f a, v2f b, v8f c) {
    // D(16x16,f32) = A(16x4,f32) x B(4x16,f32) + C
    return __builtin_amdgcn_wmma_f32_16x16x4_f32(
        false, a, false, b, (short)0, c, false, false);
}

__global__ __launch_bounds__(128) void fcm_kernel(const float* __restrict__ x,
                                                  float* __restrict__ out) {
    __shared__ float sPatch[NWAVES][PATCH_H * PSTR];   // 6336 B / wave
    __shared__ float sHg[NWAVES][16 * CSTR];           // 2816 B / wave (col-major)
    __shared__ float sHb[NWAVES][16 * CSTR];
    __shared__ float sHs[NWAVES][16 * CSTR];

    const int tid  = threadIdx.x;
    const int wv   = tid >> 5;
    const int lane = tid & 31;
    const int half = lane >> 4;      // 0: lanes 0-15, 1: lanes 16-31
    const int nr   = lane & 15;      // M/N index within 16

    // 1024 planes x 8 row-tiles(32) x 16 col-tiles(16).
    // tileId is wave-uniform; force it into an SGPR so the compiler emits
    // scalar branches (no EXEC masking around WMMA) and scalar address bases.
    const int tileId = __builtin_amdgcn_readfirstlane(blockIdx.x * NWAVES + wv);
    const int c0 = (tileId & 15) * TILE_W;
    const int r0 = ((tileId >> 4) & 7) * TILE_H;
    const int plane = tileId >> 7;

    const float* __restrict__ xp = x + (size_t)plane * (IMG_H * IMG_W);
    float* __restrict__ op = out + (size_t)plane * (IMG_H * IMG_W);
    float* P  = sPatch[wv];
    float* HG = sHg[wv];
    float* HB = sHb[wv];
    float* HS = sHs[wv];

    // ---- per-lane Toeplitz chunk constants (serve horizontal-B AND vertical-A
    //      by the f32 WMMA operand layouts: v0 <- K=4c+2*half, v1 <- K+1) ----
    v2f Tg[7], Tb[7];
    #pragma unroll
    for (int c = 0; c < 7; ++c) {
        const int k0 = 4 * c + 2 * half;
        #pragma unroll
        for (int e = 0; e < 2; ++e) {
            const int d = k0 + e - nr;                 // tap index w[d]
            const bool in = (d >= 0) && (d <= 12);
            const float dd = (float)(d - 6);
            Tg[c][e] = in ? __expf(-0.125f * dd * dd) * 0.19967563f : 0.0f;
            Tb[c][e] = in ? (1.0f / 13.0f) : 0.0f;
        }
    }

    // ---- stage 0: fill zero-padded 44x32 patch with b128 loads/stores ----
    {
        const int pr  = lane >> 3;          // row within quad of rows
        const int pc4 = (lane & 7) * 4;     // float4 column
        const int gc4 = c0 - PADL + pc4;    // 16B-aligned global col
        const bool cin = (gc4 >= 0) && (gc4 < IMG_W);
        #pragma unroll
        for (int it = 0; it < 11; ++it) {
            const int r  = it * 4 + pr;
            const int gr = r0 - PADT + r;
            v4f v = {0.0f, 0.0f, 0.0f, 0.0f};
            if (cin && gr >= 0 && gr < IMG_H)
                v = *(const v4f*)(xp + gr * IMG_W + gc4);
            *(v4f*)(P + r * PSTR + pc4) = v;
        }
    }
    __syncthreads();

    // ---- stage 1: horizontal passes (rows 0..43 in groups {0,16,28}) ----
    const bool cEdge = (c0 == 0) || (c0 == 240);     // scalar condition
    int pg[14];
    #pragma unroll
    for (int i = 0; i < 14; ++i) pg[i] = 0;
    if (cEdge) {
        #pragma unroll
        for (int c = 0; c < 7; ++c) {
            const int k0 = 4 * c + 2 * half;
            pg[2 * c]     = refl256(c0 - PADT + k0)     - c0 + PADL;
            pg[2 * c + 1] = refl256(c0 - PADT + k0 + 1) - c0 + PADL;
        }
    }

    #pragma unroll
    for (int g = 0; g < 3; ++g) {
        const int rg = (g == 0) ? 0 : ((g == 1) ? 16 : 28);
        const int rowbase = (rg + nr) * PSTR;
        v8f hg = {}; v8f hb = {}; v8f hs = {};
        if (cEdge) {
            #pragma unroll
            for (int c = 0; c < 7; ++c) {
                const int k0 = 4 * c + 2 * half;
                const v2f Ab = *(const v2f*)(P + rowbase + k0 + 2);
                const v2f As = {Ab[0] * Ab[0], Ab[1] * Ab[1]};
                const v2f Ag = {P[rowbase + pg[2 * c]], P[rowbase + pg[2 * c + 1]]};
                hb = wmma4(Ab, Tb[c], hb);
                hs = wmma4(As, Tb[c], hs);
                hg = wmma4(Ag, Tg[c], hg);
            }
        } else {
            #pragma unroll
            for (int c = 0; c < 7; ++c) {
                const int k0 = 4 * c + 2 * half;
                const v2f Ab = *(const v2f*)(P + rowbase + k0 + 2);
                const v2f As = {Ab[0] * Ab[0], Ab[1] * Ab[1]};
                hb = wmma4(Ab, Tb[c], hb);
                hs = wmma4(As, Tb[c], hs);
                hg = wmma4(Ab, Tg[c], hg);   // interior: gauss A == box A
            }
        }
        // column-major stores: 8 consecutive rows per lane -> 2x ds_store_b128
        const int cb = nr * CSTR + rg + 8 * half;
        v4f a0 = {hg[0], hg[1], hg[2], hg[3]}; v4f a1 = {hg[4], hg[5], hg[6], hg[7]};
        *(v4f*)(&HG[cb]) = a0; *(v4f*)(&HG[cb + 4]) = a1;
        v4f b0 = {hb[0], hb[1], hb[2], hb[3]}; v4f b1 = {hb[4], hb[5], hb[6], hb[7]};
        *(v4f*)(&HB[cb]) = b0; *(v4f*)(&HB[cb + 4]) = b1;
        v4f s0 = {hs[0], hs[1], hs[2], hs[3]}; v4f s1 = {hs[4], hs[5], hs[6], hs[7]};
        *(v4f*)(&HS[cb]) = s0; *(v4f*)(&HS[cb + 4]) = s1;
    }
    __syncthreads();

    // ---- row-edge fixup: write reflected rows into gauss intermediate so the
    //      vertical gauss pass uses the same direct indices as the box pass ----
    if (r0 == 0) {                                  // scalar branch
        if (lane < 16) {
            #pragma unroll
            for (int i = 0; i < 6; ++i)
                HG[lane * CSTR + i] = HG[lane * CSTR + (12 - i)];
        }
    } else if (r0 == 224) {                         // scalar branch
        if (lane < 16) {
            #pragma unroll
            for (int i = 38; i < 44; ++i)
                HG[lane * CSTR + i] = HG[lane * CSTR + (74 - i)];
        }
    }
    __syncthreads();

    // ---- stage 2: vertical passes + fused pointwise, two 16-row tiles ----
    #pragma unroll
    for (int t = 0; t < 2; ++t) {
        v8f vg = {}; v8f vb = {}; v8f vs = {};
        const int cb = nr * CSTR + 16 * t;
        #pragma unroll
        for (int c = 0; c < 7; ++c) {
            const int k0 = 4 * c + 2 * half;
            const v2f Bg = *(const v2f*)(&HG[cb + k0]);
            const v2f Bb = *(const v2f*)(&HB[cb + k0]);
            const v2f Bs = *(const v2f*)(&HS[cb + k0]);
            vg = wmma4(Tg[c], Bg, vg);
            vb = wmma4(Tb[c], Bb, vb);
            vs = wmma4(Tb[c], Bs, vs);
        }
        #pragma unroll
        for (int v = 0; v < 8; ++v) {
            const int m = v + 8 * half;                 // output row within 16-tile
            const float xc = P[(16 * t + m + PADT) * PSTR + nr + PADL];
            const float G = vg[v];                      // gaussian mean
            const float M = vb[v];                      // box mean
            const float S = vs[v];                      // box mean of squares
            const float sqm  = fmaxf(M * M, 1e-7f);
            const float stdl = sqrtf(fabsf(S - sqm) + 1e-7f);
            const float gn   = (xc - G) / (fabsf(G) + 1e-7f);
            const float num  = (gn > 0.0f) ? sqrtf(gn) : 0.0f;   // msk*|gn|^0.5
            const float ln   = stdl / (fabsf(M) + 1e-7f);
            const float den  = 1.0f + sqrtf(ln);                 // 1 + ln^0.5
            __builtin_nontemporal_store(num / den,
                op + (r0 + 16 * t + m) * IMG_W + (c0 + nr));
        }
    }
}

extern "C" void kernel_launch(void* const* d_in, const int* in_sizes, int n_in,
                              void* d_out, int out_size, void* d_ws, size_t ws_size,
                              hipStream_t stream) {
    (void)in_sizes; (void)n_in; (void)d_ws; (void)ws_size; (void)out_size;
    const float* x = (const float*)d_in[0];
    float* out = (float*)d_out;
    // 16*64 planes * 8 row-tiles * 16 col-tiles = 131072 tiles, 4 per block
    fcm_kernel<<<dim3(32768), dim3(128), 0, stream>>>(x, out);
}